// RMB_1812476199094
// MI455X (gfx1250) — compile-verified
//
#include <hip/hip_runtime.h>
#include <cstdint>

// Problem constants (from reference)
#define B_    8
#define H_    64
#define W_    64
#define DM_   96
#define DI_   192        // 2.0 * DM
#define DS_   16         // state dim N
#define DR_   6          // dt rank
#define K_    4          // scan directions
#define L_    4096       // H*W
#define M_    (B_*L_)    // 32768 rows for the big GEMMs
#define NIN_  (2*DI_)    // 384
#define CDIM_ (DR_+2*DS_) // 38

typedef float v2f __attribute__((ext_vector_type(2)));
typedef float v8f __attribute__((ext_vector_type(8)));

__device__ __forceinline__ float silu_f(float x)     { return x / (1.f + __expf(-x)); }
__device__ __forceinline__ float softplus_f(float x) { return (x > 20.f) ? x : log1pf(__expf(x)); }

// ---------------------------------------------------------------------------
// WMMA f32 16x16x4 GEMM, NTP adjacent N-tiles per wave (A-fragment reused NTP
// times per K-step).  Fragment layouts per ISA 7.12.2:
//   A (16x4 f32): lanes 0-15 rows M, VGPR0/1 = K{0,1}; lanes 16-31 K{2,3}.
//   B (4x16):     row-striped across lanes; lane halves carry K{0,1}/{2,3}.
//   C/D:          VGPR r -> M = r (+8 upper lane half), N = lane&15.
// ---------------------------------------------------------------------------
template<int KD, int ND, int NTP>
__device__ __forceinline__ void gemm_tiles_f32(const float* __restrict__ A,
                                               const float* __restrict__ Bm,
                                               float* __restrict__ C,
                                               int mt, int ntg, int lane) {
  const int row  = mt*16 + (lane & 15);
  const int col0 = ntg*NTP*16 + (lane & 15);
  const int koff = (lane >> 4) * 2;
  v8f acc[NTP] = {};
  for (int k0 = 0; k0 < KD; k0 += 4) {
    const int kd = k0 + koff;
    v2f a;
    a.x = A[(size_t)row*KD + kd];
    a.y = A[(size_t)row*KD + kd + 1];
#pragma unroll
    for (int t = 0; t < NTP; ++t) {
      v2f b;
      b.x = Bm[(size_t)kd*ND     + col0 + t*16];
      b.y = Bm[(size_t)(kd+1)*ND + col0 + t*16];
      acc[t] = __builtin_amdgcn_wmma_f32_16x16x4_f32(false, a, false, b,
                                                     (short)0, acc[t], false, false);
    }
  }
  const int rbase = mt*16 + ((lane >> 4) << 3);
#pragma unroll
  for (int t = 0; t < NTP; ++t)
#pragma unroll
    for (int r = 0; r < 8; ++r)
      C[(size_t)(rbase + r)*ND + col0 + t*16] = acc[t][r];
}

// ---------------------------------------------------------------------------
// Kernel 1: in-projection  xz[M,384] = x[M,96] @ W_in[96,384]
// 16x64 per wave (4 N-tiles), 2048 x 6 wave-tiles.
// ---------------------------------------------------------------------------
__global__ void __launch_bounds__(256)
k_inproj(const float* __restrict__ x, const float* __restrict__ Wm,
         float* __restrict__ xz) {
  const int lane = threadIdx.x & 31;
  const int wtile = blockIdx.x * 8 + (threadIdx.x >> 5);
  const int NG = NIN_/(16*4);             // 6 N-groups
  gemm_tiles_f32<DM_, NIN_, 4>(x, Wm, xz, wtile / NG, wtile % NG, lane);
}

// ---------------------------------------------------------------------------
// Kernel 2: depthwise 3x3 conv (SAME) + bias + SiLU.
// Reads xc half of xz (NHWC rows); writes xconv in (B, L, DI) layout so the
// scan's 192 channel lanes read one contiguous row per step (L2-resident).
// ---------------------------------------------------------------------------
__global__ void __launch_bounds__(256)
k_conv(const float* __restrict__ xz, const float* __restrict__ cw,
       const float* __restrict__ cb, float* __restrict__ xcv) {
  const long t = (long)blockIdx.x * blockDim.x + threadIdx.x;
  if (t >= (long)B_*L_*DI_) return;
  const int d = (int)(t % DI_);
  const int l = (int)((t / DI_) % L_);
  const int b = (int)(t / ((long)DI_*L_));
  const int hh = l >> 6, ww = l & 63;
  float acc = cb[d];
#pragma unroll
  for (int dh = -1; dh <= 1; ++dh)
#pragma unroll
    for (int dw = -1; dw <= 1; ++dw) {
      const int h2 = hh + dh, w2 = ww + dw;
      if (h2 >= 0 && h2 < H_ && w2 >= 0 && w2 < W_)
        acc = fmaf(xz[((size_t)b*L_ + h2*W_ + w2)*NIN_ + d],
                   cw[d*9 + (dh+1)*3 + (dw+1)], acc);
    }
  xcv[((size_t)b*L_ + l)*DI_ + d] = silu_f(acc);
}

// ---------------------------------------------------------------------------
// Kernel 3: x_dbl projection with WMMA.
//   c[b,k,l,0:38] = x_proj_w[k] (38x192) @ xconv[b, scan_ids[k,l], :]
// All 3 row-tiles per wave so the gathered B-fragment is loaded ONCE per
// K-step.  Rows 0..5 = dt_r, 6..21 = B, 22..37 = C (delta recomputed in scan).
// ---------------------------------------------------------------------------
__global__ void __launch_bounds__(256)
k_xdbl(const float* __restrict__ xcv, const float* __restrict__ xpw,
       const int* __restrict__ scan_ids, float* __restrict__ xdbl) {
  const int lane  = threadIdx.x & 31;
  const int wtile = blockIdx.x * 8 + (threadIdx.x >> 5);
  const int lt = wtile % (L_/16);
  const int bk = wtile / (L_/16);
  const int b = bk >> 2, k = bk & 3;
  const int koff = (lane >> 4) * 2;
  const int mrow = lane & 15;
  const float mmask2 = (mrow < CDIM_ - 32) ? 1.f : 0.f;  // row-tile 2: rows 32..47
  const int l = lt*16 + (lane & 15);
  const int j = scan_ids[k*L_ + l];       // per-column gather index
  const float* A0 = xpw + (size_t)k*CDIM_*DI_;
  const float* Bp = xcv + ((size_t)b*L_ + j)*DI_;
  v8f acc[3] = {};
  for (int k0 = 0; k0 < DI_; k0 += 4) {
    const int kd = k0 + koff;
    v2f bf;
    bf.x = Bp[kd];
    bf.y = Bp[kd + 1];
#pragma unroll
    for (int rt = 0; rt < 3; ++rt) {
      const int m  = rt*16 + mrow;
      const int mc = (rt < 2) ? m : (mrow < CDIM_ - 32 ? m : 0);
      const float msk = (rt < 2) ? 1.f : mmask2;
      v2f a;
      a.x = A0[(size_t)mc*DI_ + kd]     * msk;
      a.y = A0[(size_t)mc*DI_ + kd + 1] * msk;
      acc[rt] = __builtin_amdgcn_wmma_f32_16x16x4_f32(false, a, false, bf,
                                                      (short)0, acc[rt], false, false);
    }
  }
  float* outp = xdbl + ((size_t)bk*L_ + l)*CDIM_;
  const int rb = (lane >> 4) << 3;
#pragma unroll
  for (int rt = 0; rt < 3; ++rt)
#pragma unroll
    for (int r = 0; r < 8; ++r) {
      const int mm = rt*16 + rb + r;
      if (mm < CDIM_) outp[mm] = acc[rt][r];
    }
}

// ---------------------------------------------------------------------------
// Kernel 4: selective scan. One block per (b,k); lane = channel d.
// h[16] state in VGPRs; (dt,B,C) stream staged into LDS in 32-step chunks via
// GLOBAL_LOAD_ASYNC_TO_LDS_B128 (ASYNCcnt, no VGPR round-trip); u gathered
// from L2-resident xconv; y scattered with fp32 atomic-add through scan_ids
// (replaces the inverse-permute gather).
// ---------------------------------------------------------------------------
__global__ void __launch_bounds__(DI_)
k_scan(const float* __restrict__ xcv, const float* __restrict__ xdbl,
       const int* __restrict__ scan_ids,
       const float* __restrict__ dt_w, const float* __restrict__ dt_b,
       const float* __restrict__ A_logs, const float* __restrict__ Dsp,
       float* __restrict__ yacc) {
  const int bk = blockIdx.x;
  const int b = bk >> 2, k = bk & 3;
  const int d = threadIdx.x;
  __shared__ __align__(16) float sc[32 * CDIM_];   // 4864 B, 16B-aligned chunks
  __shared__ int sj[32];

  float dtw[DR_];
#pragma unroll
  for (int r = 0; r < DR_; ++r) dtw[r] = dt_w[((size_t)k*DI_ + d)*DR_ + r];
  const float dtb = dt_b[k*DI_ + d];
  const float Dd  = Dsp[k*DI_ + d];
  float an[DS_];
#pragma unroll
  for (int n = 0; n < DS_; ++n) an[n] = -__expf(A_logs[((size_t)k*DI_ + d)*DS_ + n]);
  float h[DS_];
#pragma unroll
  for (int n = 0; n < DS_; ++n) h[n] = 0.f;

  const float* cbase = xdbl + (size_t)bk*L_*CDIM_;
  const int*   jbase = scan_ids + k*L_;

  for (int c0 = 0; c0 < L_; c0 += 32) {
    if (c0 + 32 < L_)  // pull next chunk toward L2/L0 while we compute
      __builtin_prefetch(cbase + (size_t)(c0 + 32)*CDIM_ + threadIdx.x*8, 0, 1);
    // Async DMA chunk (32 steps x 38 floats = 4864 B) into LDS, 16B packets.
    const float* src = cbase + (size_t)c0*CDIM_;
    for (int i4 = threadIdx.x*4; i4 < 32*CDIM_; i4 += DI_*4) {
      const uint32_t ldsa = (uint32_t)(uintptr_t)(&sc[i4]);  // LDS = flat[31:0]
      asm volatile("global_load_async_to_lds_b128 %0, %1, off"
                   :: "v"(ldsa), "v"(src + i4) : "memory");
    }
    if (threadIdx.x < 32) sj[threadIdx.x] = jbase[c0 + threadIdx.x];
    asm volatile("s_wait_asynccnt 0x0" ::: "memory");
    __syncthreads();

    for (int s = 0; s < 32; ++s) {
      const float* c = &sc[s*CDIM_];
      const int j = sj[s];
      const float u = xcv[((size_t)b*L_ + j)*DI_ + d];
      float pre = dtb;
#pragma unroll
      for (int r = 0; r < DR_; ++r) pre = fmaf(dtw[r], c[r], pre);
      const float delta = softplus_f(pre);
      const float du = delta * u;
      float yv = 0.f;
#pragma unroll
      for (int n = 0; n < DS_; ++n) {
        const float dA = __expf(delta * an[n]);
        h[n] = fmaf(h[n], dA, du * c[DR_ + n]);
        yv = fmaf(h[n], c[DR_ + DS_ + n], yv);
      }
      yv = fmaf(Dd, u, yv);
      (void)__hip_atomic_fetch_add(&yacc[((size_t)b*L_ + j)*DI_ + d], yv,
                                   __ATOMIC_RELAXED, __HIP_MEMORY_SCOPE_AGENT);
    }
    __syncthreads();
  }
}

// ---------------------------------------------------------------------------
// Kernel 5: LayerNorm over DI + SiLU(z) gating. One block per (b, spatial).
// ---------------------------------------------------------------------------
__global__ void __launch_bounds__(DI_)
k_ln(const float* __restrict__ yacc, const float* __restrict__ xz,
     const float* __restrict__ g, const float* __restrict__ bt,
     float* __restrict__ yln) {
  const int bj = blockIdx.x;
  const int d  = threadIdx.x;
  __shared__ float s1[DI_], s2[DI_];
  const float v = yacc[(size_t)bj*DI_ + d];
  s1[d] = v; s2[d] = v*v;
  __syncthreads();
  if (d < 64) { s1[d] += s1[d+64] + s1[d+128]; s2[d] += s2[d+64] + s2[d+128]; }
  __syncthreads();
  for (int st = 32; st > 0; st >>= 1) {
    if (d < st) { s1[d] += s1[d+st]; s2[d] += s2[d+st]; }
    __syncthreads();
  }
  const float mu   = s1[0] * (1.f/DI_);
  const float var  = s2[0] * (1.f/DI_) - mu*mu;
  const float rstd = rsqrtf(var + 1e-5f);
  const float z    = xz[(size_t)bj*NIN_ + DI_ + d];
  yln[(size_t)bj*DI_ + d] = ((v - mu)*rstd*g[d] + bt[d]) * silu_f(z);
}

// ---------------------------------------------------------------------------
// Kernel 6: out-projection  out[M,96] = yln[M,192] @ W_out[192,96]
// Full 16x96 row strip per wave (6 N-tiles, A-fragment reused 6x).
// ---------------------------------------------------------------------------
__global__ void __launch_bounds__(256)
k_outproj(const float* __restrict__ a, const float* __restrict__ Wm,
          float* __restrict__ o) {
  const int lane = threadIdx.x & 31;
  const int mt   = blockIdx.x * 8 + (threadIdx.x >> 5);
  gemm_tiles_f32<DI_, DM_, 6>(a, Wm, o, mt, 0, lane);
}

// ---------------------------------------------------------------------------
extern "C" void kernel_launch(void* const* d_in, const int* in_sizes, int n_in,
                              void* d_out, int out_size, void* d_ws, size_t ws_size,
                              hipStream_t stream) {
  (void)in_sizes; (void)n_in; (void)out_size; (void)ws_size;
  const float* x        = (const float*)d_in[0];
  const int*   scan_ids = (const int*)  d_in[1];
  /* d_in[2] inv_ids unused: scatter formulation */
  const float* W_in     = (const float*)d_in[3];
  const float* conv_w   = (const float*)d_in[4];
  const float* conv_b   = (const float*)d_in[5];
  const float* x_proj_w = (const float*)d_in[6];
  const float* dt_w     = (const float*)d_in[7];
  const float* dt_b     = (const float*)d_in[8];
  const float* A_logs   = (const float*)d_in[9];
  const float* Ds       = (const float*)d_in[10];
  const float* ln_g     = (const float*)d_in[11];
  const float* ln_b     = (const float*)d_in[12];
  const float* W_out    = (const float*)d_in[13];
  float* out = (float*)d_out;

  char*  ws   = (char*)d_ws;
  size_t off  = 0;
  float* xz   = (float*)(ws + off); off += (size_t)M_*NIN_*sizeof(float);   // 50.3 MB
  float* xcv  = (float*)(ws + off); off += (size_t)M_*DI_*sizeof(float);    // 25.2 MB
  float* xdbl = (float*)(ws + off); off += (size_t)B_*K_*L_*CDIM_*sizeof(float); // 19.9 MB
  float* yacc = (float*)(ws + off); off += (size_t)M_*DI_*sizeof(float);    // 25.2 MB
  float* yln  = (float*)(ws + off);                                          // 25.2 MB

  // 1) in-projection GEMM (WMMA f32): 2048 M-tiles x 6 N-groups, 8 waves/block
  k_inproj<<<(M_/16)*(NIN_/64)/8, 256, 0, stream>>>(x, W_in, xz);
  // 2) depthwise conv + SiLU
  {
    const long total = (long)B_*L_*DI_;
    k_conv<<<(unsigned)((total + 255)/256), 256, 0, stream>>>(xz, conv_w, conv_b, xcv);
  }
  // 3) x_dbl projection (WMMA f32, gathered B loaded once per K-step)
  k_xdbl<<<(B_*K_*(L_/16))/8, 256, 0, stream>>>(xcv, x_proj_w, scan_ids, xdbl);
  // 4) zero scatter accumulator, then selective scan (async-LDS staging)
  (void)hipMemsetAsync(yacc, 0, (size_t)M_*DI_*sizeof(float), stream);
  k_scan<<<B_*K_, DI_, 0, stream>>>(xcv, xdbl, scan_ids,
                                    dt_w, dt_b, A_logs, Ds, yacc);
  // 5) LayerNorm + SiLU(z) gate
  k_ln<<<M_, DI_, 0, stream>>>(yacc, xz, ln_g, ln_b, yln);
  // 6) out-projection GEMM (WMMA f32)
  k_outproj<<<(M_/16)/8, 256, 0, stream>>>(yln, W_out, out);
}